// Rule_15221364097272
// MI455X (gfx1250) — compile-verified
//
#include <hip/hip_runtime.h>
#include <stdint.h>

// Problem dims (fixed by the reference setup)
#define BB 2
#define HH 2048
#define WW 2048
#define TILE 32
#define HALO 2
#define LW   (TILE + 2 * HALO)   // 36 staged rows/cols
#define LSTR 40                  // padded LDS row stride (floats), 16B-aligned
#define NTHREADS 256
#define NSTAGE ((LW * LW + NTHREADS - 1) / NTHREADS)  // 6

#define AS1 __attribute__((address_space(1)))
#define AS3 __attribute__((address_space(3)))

#if defined(__has_builtin)
#if __has_builtin(__builtin_amdgcn_global_load_async_to_lds_b32)
#define USE_ASYNC_BUILTIN 1
#endif
#if __has_builtin(__builtin_amdgcn_s_wait_asynccnt)
#define USE_WAIT_BUILTIN 1
#endif
#endif

// CDNA5 async global->LDS DMA (one f32 per lane, per-lane global AND LDS addr).
__device__ __forceinline__ void async_copy_f32(const float* g, float* l) {
#if defined(USE_ASYNC_BUILTIN)
  __builtin_amdgcn_global_load_async_to_lds_b32(
      (AS1 int*)(int*)(void*)(const_cast<float*>(g)),
      (AS3 int*)(int*)(void*)l, 0, 0);
#else
  unsigned loff = (unsigned)(__SIZE_TYPE__)(AS3 void*)l;
  asm volatile("global_load_async_to_lds_b32 %0, %1, off"
               :: "v"(loff), "v"(g) : "memory");
#endif
}

__device__ __forceinline__ void wait_async0() {
#if defined(USE_WAIT_BUILTIN)
  __builtin_amdgcn_s_wait_asynccnt(0);
#else
  asm volatile("s_wait_asynccnt 0x0" ::: "memory");
#endif
}

__global__ __launch_bounds__(NTHREADS) void snn_step_kernel(
    const float* __restrict__ x, const float* __restrict__ noise,
    const float* __restrict__ ei, const float* __restrict__ kern,
    const int* __restrict__ coin, float* __restrict__ out) {
  __shared__ float s_in[LW * LSTR];  // staged x0 tile, becomes s_in = x0*EI
  __shared__ float s_ei[LW * LSTR];  // staged EI tile
  __shared__ float s_k[32];          // 25 conv weights

  const int tid = threadIdx.x;
  const int bx = blockIdx.x * TILE;
  const int by = blockIdx.y * TILE;
  const int b  = blockIdx.z;
  const size_t HW = (size_t)HH * WW;

  const float* x0p = x + (size_t)b * 6 * HW;  // channel 0 of x
  const float* eip = ei + (size_t)b * HW;

  // ---- Stage halo tiles (wrap addressing) via async global->LDS ----
#pragma unroll
  for (int it = 0; it < NSTAGE; ++it) {
    const int i = tid + it * NTHREADS;
    if (i < LW * LW) {
      const int ry = i / LW;
      const int rx = i - ry * LW;
      const int gy = (by + ry - HALO) & (HH - 1);
      const int gx = (bx + rx - HALO) & (WW - 1);
      const size_t go = (size_t)gy * WW + gx;
      async_copy_f32(x0p + go, &s_in[ry * LSTR + rx]);
      async_copy_f32(eip + go, &s_ei[ry * LSTR + rx]);
    }
  }
  if (tid < 25) s_k[tid] = kern[tid];
  wait_async0();
  __syncthreads();

  // ---- Fuse: s_in = x0 * EI (in place, disjoint indices per thread) ----
#pragma unroll
  for (int it = 0; it < NSTAGE; ++it) {
    const int i = tid + it * NTHREADS;
    if (i < LW * LW) {
      const int ry = i / LW;
      const int rx = i - ry * LW;
      s_in[ry * LSTR + rx] *= s_ei[ry * LSTR + rx];
    }
  }
  __syncthreads();

  // ---- Conv weights into registers (broadcast LDS reads) ----
  float w[25];
#pragma unroll
  for (int t = 0; t < 25; ++t) w[t] = s_k[t];

  // Thread -> 4 consecutive pixels: 8 threads per row, 32 rows
  const int lx = (tid & 7) * 4;
  const int ly = tid >> 3;

  // ---- 5x5 convolution from LDS with b128 row reads ----
  float acc[4] = {0.f, 0.f, 0.f, 0.f};
#pragma unroll
  for (int ky = 0; ky < 5; ++ky) {
    const float4 ra = *(const float4*)&s_in[(ly + ky) * LSTR + lx];
    const float4 rb = *(const float4*)&s_in[(ly + ky) * LSTR + lx + 4];
    const float r[8] = {ra.x, ra.y, ra.z, ra.w, rb.x, rb.y, rb.z, rb.w};
#pragma unroll
    for (int kx = 0; kx < 5; ++kx) {
      const float wt = w[ky * 5 + kx];
      acc[0] = fmaf(r[kx + 0], wt, acc[0]);
      acc[1] = fmaf(r[kx + 1], wt, acc[1]);
      acc[2] = fmaf(r[kx + 2], wt, acc[2]);
      acc[3] = fmaf(r[kx + 3], wt, acc[3]);
    }
  }

  // ---- EI for this thread's 4 pixels straight from the staged LDS tile ----
  const int ic = (ly + HALO) * LSTR + (lx + HALO);
  const float Ii[4] = {s_ei[ic + 0], s_ei[ic + 1], s_ei[ic + 2], s_ei[ic + 3]};

  // ---- Elementwise neuron update (float4 streams) ----
  const size_t row  = (size_t)(by + ly) * WW + (size_t)(bx + lx);
  const size_t xb   = (size_t)b * 6 * HW + row;
  const size_t pb   = (size_t)b * HW + row;

  const float4 V4 = *(const float4*)(x + xb + 1 * HW);
  const float4 R4 = *(const float4*)(x + xb + 2 * HW);
  const float4 A4 = *(const float4*)(x + xb + 3 * HW);
  const float4 T4 = *(const float4*)(x + xb + 4 * HW);
  const float4 E4 = *(const float4*)(x + xb + 5 * HW);
  const float4 N4 = *(const float4*)(noise + pb);
  const int4   C4 = *(const int4*)(coin + pb);

  const float Vi[4] = {V4.x, V4.y, V4.z, V4.w};
  const float Ri[4] = {R4.x, R4.y, R4.z, R4.w};
  const float Ai[4] = {A4.x, A4.y, A4.z, A4.w};
  const float Ti[4] = {T4.x, T4.y, T4.z, T4.w};
  const float Ei[4] = {E4.x, E4.y, E4.z, E4.w};
  const float Ni[4] = {N4.x, N4.y, N4.z, N4.w};
  const int   Ci[4] = {C4.x, C4.y, C4.z, C4.w};

  float sp[4], Vo[4], Ro[4], Ao[4], To[4], Eo[4];
#pragma unroll
  for (int j = 0; j < 4; ++j) {
    float v = Vi[j] + Ni[j];
    v = fmaf(0.05f, acc[j] - v, v);                 // V += 0.05*(-V + I)
    float r = Ri[j] + (Ci[j] ? 1.0f : -1.0f);       // R += coin*2 - 1
    const float spike =
        ((v > Ti[j]) && (r > 5.0f) && (Ei[j] > 0.1f)) ? 1.0f : 0.0f;
    float e = Ei[j] + 0.0058f * (1.0f - Ei[j]) - spike * 0.1f;
    r = (r + 1.0f) * (1.0f - spike);
    float a = Ai[j] - Ai[j] * (1.0f / 200.0f) + spike;
    float t = Ti[j] + 0.01f * (a - 5.0f);
    t = fmaxf(t, 0.0f);
    v = v * (1.0f - spike);
    sp[j] = spike; Vo[j] = v; Ro[j] = r; Ao[j] = a; To[j] = t; Eo[j] = e;
  }

  float* ob = out + (size_t)b * 7 * HW + row;
  *(float4*)(ob + 0 * HW) = make_float4(sp[0], sp[1], sp[2], sp[3]);
  *(float4*)(ob + 1 * HW) = make_float4(Vo[0], Vo[1], Vo[2], Vo[3]);
  *(float4*)(ob + 2 * HW) = make_float4(Ro[0], Ro[1], Ro[2], Ro[3]);
  *(float4*)(ob + 3 * HW) = make_float4(Ao[0], Ao[1], Ao[2], Ao[3]);
  *(float4*)(ob + 4 * HW) = make_float4(To[0], To[1], To[2], To[3]);
  *(float4*)(ob + 5 * HW) = make_float4(Eo[0], Eo[1], Eo[2], Eo[3]);
  *(float4*)(ob + 6 * HW) = make_float4(Ii[0], Ii[1], Ii[2], Ii[3]);
}

extern "C" void kernel_launch(void* const* d_in, const int* in_sizes, int n_in,
                              void* d_out, int out_size, void* d_ws,
                              size_t ws_size, hipStream_t stream) {
  (void)in_sizes; (void)n_in; (void)out_size; (void)d_ws; (void)ws_size;
  const float* x     = (const float*)d_in[0];
  const float* noise = (const float*)d_in[1];
  const float* ei    = (const float*)d_in[2];
  const float* kern  = (const float*)d_in[3];
  const int*   coin  = (const int*)d_in[4];
  float* out = (float*)d_out;

  dim3 grid(WW / TILE, HH / TILE, BB);
  snn_step_kernel<<<grid, NTHREADS, 0, stream>>>(x, noise, ei, kern, coin, out);
}